// RocketFeatures_89309549953858
// MI455X (gfx1250) — compile-verified
//
#include <hip/hip_runtime.h>
#include <hip/hip_bf16.h>

typedef float v2f __attribute__((ext_vector_type(2)));
typedef float v8f __attribute__((ext_vector_type(8)));

// Problem constants (reference): N=512, C=12, L=8192, PER=5 filters/group,
// 3 kernel sizes x 6 dilations = 18 groups, 10 features/group -> 180 features.
#define RN   512
#define RC   12
#define RL   8192
#define RPER 5
#define RNF  180

// Fused im2col K-dimension over all three kernel sizes (same dilation):
//   kidx in [0,84)    -> KS=7,  c=kidx/7,        k=kidx%7,        kc=3
//   kidx in [84,192)  -> KS=9,  c=(kidx-84)/9,   k=(kidx-84)%9,   kc=4
//   kidx in [192,324) -> KS=11, c=(kidx-192)/11, k=(kidx-192)%11, kc=5
// Segment boundaries (84, 192) are even multiples of 4: K-steps (width 4)
// and K-pairs (width 2) never straddle a segment.
// 15 real filters (5 per kernel size) in M=16.
#define KD7   84
#define KD9   108
#define KD11  132
#define KDIM  324
#define NSTEP 81

struct CK { int c; int off; };
static __device__ __forceinline__ CK ck_of(int kidx) {
    if (kidx < KD7)         return { kidx / 7,          (kidx % 7) - 3 };
    if (kidx < KD7 + KD9)   return { (kidx - 84) / 9,   ((kidx - 84) % 9) - 4 };
    return                         { (kidx - 192) / 11, ((kidx - 192) % 11) - 5 };
}

// Fused block-diagonal weight value W[m][kidx] (m in [0,16), zero padding).
static __device__ __forceinline__ float w_of(int kidx, int m, int dIdx,
                                             const float* __restrict__ W7,
                                             const float* __restrict__ W9,
                                             const float* __restrict__ W11) {
    if (kidx < KD7) {
        return (m < 5) ? W7[(size_t)(dIdx * RPER + m) * KD7 + kidx] : 0.0f;
    } else if (kidx < KD7 + KD9) {
        return (m >= 5 && m < 10) ? W9[(size_t)(dIdx * RPER + m - 5) * KD9 + (kidx - KD7)] : 0.0f;
    }
    return (m >= 10 && m < 15) ? W11[(size_t)(dIdx * RPER + m - 10) * KD11 + (kidx - KD7 - KD9)] : 0.0f;
}

// One block per (sample, dilation). Computes all 15 filters (3 kernel sizes
// x 5 filters) via fp32 WMMA (M=16 filters, K=324 fused, N=16 positions),
// then ppv (fraction > 0) and max over L, written to feat[n][180].
__launch_bounds__(256)
__global__ void rocket_conv_fused(const float* __restrict__ x,
                                  const float* __restrict__ W7,
                                  const float* __restrict__ b7,
                                  const float* __restrict__ W9,
                                  const float* __restrict__ b9,
                                  const float* __restrict__ W11,
                                  const float* __restrict__ b11,
                                  float* __restrict__ feat)
{
    constexpr int CH   = 512;           // positions per LDS chunk
    constexpr int HMAX = 160;           // worst-case halo: (11/2)*32
    constexpr int SPAN = CH + 2 * HMAX; // 832 floats per channel
    constexpr int TSTR = 8 * 16;        // float stride between a wave's tiles

    const int nIdx = blockIdx.x / 6;
    const int dIdx = blockIdx.x % 6;
    const int dil  = 1 << dIdx;         // dilations 1,2,4,8,16,32

    // A fragments in pair order: wfrag[j*16+m] = { W[m][2j], W[m][2j+1] }.
    __shared__ v2f   wfrag[(KDIM / 2) * 16];   // 20,736 B
    __shared__ float xs[RC * SPAN];            // 39,936 B
    __shared__ int2  boffp[KDIM / 2];          // 1,296 B: {c*SPAN + off*dil} pairs
    __shared__ float pcnt[8][16];
    __shared__ float pmx[8][16];

    const int tid = threadIdx.x;

    for (int i = tid; i < (KDIM / 2) * 16; i += 256) {
        const int j = i >> 4;       // K-pair index
        const int m = i & 15;       // filter row
        v2f wv;
        wv.x = w_of(2 * j + 0, m, dIdx, W7, W9, W11);
        wv.y = w_of(2 * j + 1, m, dIdx, W7, W9, W11);
        wfrag[i] = wv;
    }
    // Branch/div-heavy index math runs exactly once per block, into LDS.
    for (int j = tid; j < KDIM / 2; j += 256) {
        const CK p0 = ck_of(2 * j + 0);
        const CK p1 = ck_of(2 * j + 1);
        boffp[j] = make_int2(p0.c * SPAN + p0.off * dil,
                             p1.c * SPAN + p1.off * dil);
    }
    __syncthreads();

    const int lane  = tid & 31;
    const int wave  = tid >> 5;
    const int laneN = lane & 15;
    const bool hi   = (lane >= 16);   // lanes 0-15: K {0,1}; 16-31: K {2,3}
    const int  hsel = hi ? 1 : 0;

    // Per-lane-half bias: lanes 0-15 handle filters 0-7, lanes 16-31 filters 8-15.
    const int fbase = hi ? 8 : 0;
    float bias[8];
#pragma unroll
    for (int v = 0; v < 8; ++v) {
        const int fm = fbase + v;
        float bv = 0.0f;
        if (fm < 5)       bv = b7 [dIdx * RPER + fm];
        else if (fm < 10) bv = b9 [dIdx * RPER + fm - 5];
        else if (fm < 15) bv = b11[dIdx * RPER + fm - 10];
        bias[v] = bv;
    }

    const float* __restrict__ xrow = x + (size_t)nIdx * RC * RL;

    float cnt[8];
    float mx[8];
#pragma unroll
    for (int v = 0; v < 8; ++v) { cnt[v] = 0.0f; mx[v] = -__builtin_inff(); }

    for (int chunk = 0; chunk < RL / CH; ++chunk) {
        const int chunk0 = chunk * CH;

        __syncthreads();   // xs reuse across chunk iterations
        for (int i = tid; i < RC * SPAN; i += 256) {
            const int c = i / SPAN;
            const int p = i - c * SPAN;
            const int g = chunk0 - HMAX + p;
            xs[i] = ((unsigned)g < (unsigned)RL) ? xrow[(size_t)c * RL + g] : 0.0f;
        }
        __syncthreads();

        // This wave's 4 tiles (of 32 in the chunk) processed concurrently:
        // positions lbase + i*TSTR, i = 0..3. Shared av/bo per step,
        // 4 independent accumulators -> wmma RAW chains 4 issues apart.
        const int lbase = wave * 16 + laneN + HMAX;

        v8f acc0 = {}, acc1 = {}, acc2 = {}, acc3 = {};
#pragma unroll
        for (int s = 0; s < NSTEP; ++s) {
            const int j = 2 * s + hsel;
            const v2f  av = wfrag[j * 16 + laneN];   // ds_load_b64
            const int2 bo = boffp[j];                // ds_load_b64

            const float* __restrict__ px = &xs[bo.x + lbase];
            const float* __restrict__ py = &xs[bo.y + lbase];
            v2f b0; b0.x = px[0 * TSTR]; b0.y = py[0 * TSTR];
            v2f b1; b1.x = px[1 * TSTR]; b1.y = py[1 * TSTR];
            v2f b2; b2.x = px[2 * TSTR]; b2.y = py[2 * TSTR];
            v2f b3; b3.x = px[3 * TSTR]; b3.y = py[3 * TSTR];

            acc0 = __builtin_amdgcn_wmma_f32_16x16x4_f32(false, av, false, b0, (short)0, acc0, false, false);
            acc1 = __builtin_amdgcn_wmma_f32_16x16x4_f32(false, av, false, b1, (short)0, acc1, false, false);
            acc2 = __builtin_amdgcn_wmma_f32_16x16x4_f32(false, av, false, b2, (short)0, acc2, false, false);
            acc3 = __builtin_amdgcn_wmma_f32_16x16x4_f32(false, av, false, b3, (short)0, acc3, false, false);
        }

        // acc_i[v] = filter fbase+v at position chunk0 + wave*16 + i*TSTR + laneN.
#pragma unroll
        for (int v = 0; v < 8; ++v) {
            const float o0 = acc0[v] + bias[v];
            const float o1 = acc1[v] + bias[v];
            const float o2 = acc2[v] + bias[v];
            const float o3 = acc3[v] + bias[v];
            cnt[v] += ((o0 > 0.0f) ? 1.0f : 0.0f) + ((o1 > 0.0f) ? 1.0f : 0.0f) +
                      ((o2 > 0.0f) ? 1.0f : 0.0f) + ((o3 > 0.0f) ? 1.0f : 0.0f);
            mx[v] = fmaxf(fmaxf(fmaxf(mx[v], o0), fmaxf(o1, o2)), o3);
        }
    }

    // Reduce over the 16 positions within each lane half.
#pragma unroll
    for (int v = 0; v < 8; ++v) {
        for (int off = 8; off >= 1; off >>= 1) {
            cnt[v] += __shfl_xor(cnt[v], off, 32);
            mx[v] = fmaxf(mx[v], __shfl_xor(mx[v], off, 32));
        }
    }
    if (laneN == 0) {   // lane 0 publishes filters 0-7, lane 16 filters 8-15
#pragma unroll
        for (int v = 0; v < 8; ++v) { pcnt[wave][fbase + v] = cnt[v]; pmx[wave][fbase + v] = mx[v]; }
    }
    __syncthreads();

    if (tid < 15) {     // 15 real filters
        float c = 0.0f, m = -__builtin_inff();
#pragma unroll
        for (int w = 0; w < 8; ++w) { c += pcnt[w][tid]; m = fmaxf(m, pmx[w][tid]); }
        const int ksIdx = tid / RPER;
        const int f     = tid % RPER;
        const int base  = nIdx * RNF + (ksIdx * 6 + dIdx) * (2 * RPER) + f * 2;
        feat[base + 0] = c * (1.0f / RL);   // ppv
        feat[base + 1] = m;                 // max
    }
}

// Batch-normalize each of the 180 features over the N=512 samples.
__launch_bounds__(256)
__global__ void rocket_norm_kernel(const float* __restrict__ feat,
                                   float* __restrict__ out)
{
    const int j   = blockIdx.x;       // feature index [0,180)
    const int tid = threadIdx.x;      // 256 threads
    const int lane = tid & 31;
    const int wave = tid >> 5;

    float s = 0.0f, s2 = 0.0f;
    for (int n = tid; n < RN; n += 256) {
        const float v = feat[n * RNF + j];
        s += v;
        s2 += v * v;
    }
    for (int off = 16; off >= 1; off >>= 1) {
        s  += __shfl_xor(s,  off, 32);
        s2 += __shfl_xor(s2, off, 32);
    }

    __shared__ float ls[8], ls2[8];
    __shared__ float mean_s, rstd_s;
    if (lane == 0) { ls[wave] = s; ls2[wave] = s2; }
    __syncthreads();
    if (tid == 0) {
        float ts = 0.0f, ts2 = 0.0f;
        for (int w = 0; w < 8; ++w) { ts += ls[w]; ts2 += ls2[w]; }
        const float mean = ts * (1.0f / RN);
        const float var  = ts2 * (1.0f / RN) - mean * mean;
        mean_s = mean;
        rstd_s = rsqrtf(var + 1e-5f);
    }
    __syncthreads();

    const float mean = mean_s, rstd = rstd_s;
    for (int n = tid; n < RN; n += 256) {
        out[n * RNF + j] = (feat[n * RNF + j] - mean) * rstd;
    }
}

extern "C" void kernel_launch(void* const* d_in, const int* in_sizes, int n_in,
                              void* d_out, int out_size, void* d_ws, size_t ws_size,
                              hipStream_t stream)
{
    const float* x   = (const float*)d_in[0];
    const float* W7  = (const float*)d_in[1];
    const float* b7  = (const float*)d_in[2];
    const float* W9  = (const float*)d_in[3];
    const float* b9  = (const float*)d_in[4];
    const float* W11 = (const float*)d_in[5];
    const float* b11 = (const float*)d_in[6];

    float* feat = (float*)d_ws;   // N*180 fp32 = 368,640 B of scratch

    rocket_conv_fused<<<dim3(RN * 6), dim3(256), 0, stream>>>(
        x, W7, b7, W9, b9, W11, b11, feat);

    rocket_norm_kernel<<<dim3(RNF), dim3(256), 0, stream>>>(feat, (float*)d_out);
}